// Encoder_48301202210902
// MI455X (gfx1250) — compile-verified
//
#include <hip/hip_runtime.h>
#include <hip/hip_bf16.h>
#include <math.h>

typedef float v2f __attribute__((ext_vector_type(2)));
typedef float v8f __attribute__((ext_vector_type(8)));

#define BT 8192     // B*T rows
#define NF 128      // N features (= U)

// gfx1250 has native V_TANH_F32 (TRANS op, co-executes with VALU).
#if __has_builtin(__builtin_amdgcn_tanhf)
__device__ __forceinline__ float fast_tanh(float x) { return __builtin_amdgcn_tanhf(x); }
#else
__device__ __forceinline__ float fast_tanh(float x) { return tanhf(x); }
#endif

// sigmoid(x) = 0.5 * (1 + tanh(x/2)) -> single TRANS op instead of exp+rcp
__device__ __forceinline__ float sigmoidf_(float x) {
    return __builtin_fmaf(0.5f, fast_tanh(0.5f * x), 0.5f);
}

// ---------------------------------------------------------------------------
// Kernel A: z = x2d(8192x128) @ lstm_kernel(128x512); fused gate activations.
// Only the i / g / o gate columns are needed (f-gate * c_prev(=0) vanishes).
// Each wave computes one 16x16 tile of each of the 3 gates (shared A frag),
// then writes query = [h | c] (8192 x 256).
// ---------------------------------------------------------------------------
__global__ __launch_bounds__(128) void lstm_gemm_kernel(
    const float* __restrict__ x, const float* __restrict__ Wl,
    const float* __restrict__ bl, float* __restrict__ query)
{
    const int wave = (blockIdx.x << 2) + (threadIdx.x >> 5);
    const int lane = threadIdx.x & 31;
    const int lm   = lane & 15;            // M (for A) / N (for B) within tile
    const int kh   = (lane >> 4) << 1;     // K sub-offset: 0 (lanes 0-15) or 2

    const int ct = wave & 7;               // 8 column tiles cover 128 gate cols
    const int m0 = (wave >> 3) << 4;       // 512 row tiles * 16
    const int c0 = ct << 4;

    v8f acc_i = {}; v8f acc_g = {}; v8f acc_o = {};

    const float* arow = x + (m0 + lm) * NF;
    for (int kk = 0; kk < NF; kk += 4) {
        const int k0 = kk + kh;
        v2f a; a.x = arow[k0]; a.y = arow[k0 + 1];
        const float* w0 = Wl + (size_t)k0 * 512;
        const float* w1 = Wl + (size_t)(k0 + 1) * 512;
        v2f bi, bg, bo;
        bi.x = w0[c0 + lm];       bi.y = w1[c0 + lm];        // i-gate cols [0,128)
        bg.x = w0[c0 + 256 + lm]; bg.y = w1[c0 + 256 + lm];  // g-gate cols [256,384)
        bo.x = w0[c0 + 384 + lm]; bo.y = w1[c0 + 384 + lm];  // o-gate cols [384,512)
        acc_i = __builtin_amdgcn_wmma_f32_16x16x4_f32(false, a, false, bi, (short)0, acc_i, false, false);
        acc_g = __builtin_amdgcn_wmma_f32_16x16x4_f32(false, a, false, bg, (short)0, acc_g, false, false);
        acc_o = __builtin_amdgcn_wmma_f32_16x16x4_f32(false, a, false, bo, (short)0, acc_o, false, false);
    }

    // C/D layout: VGPR r -> M = r (lanes 0-15) or r+8 (lanes 16-31), N = lane%16
    const int rbase = m0 + ((lane >> 4) << 3);
    const int col   = c0 + lm;
    const float b_i = bl[col];
    const float b_g = bl[col + 256];
    const float b_o = bl[col + 384];
#pragma unroll
    for (int r = 0; r < 8; ++r) {
        const int row = rbase + r;
        const float iv = acc_i[r] + b_i;
        const float gv = acc_g[r] + b_g;
        const float ov = acc_o[r] + b_o;
        const float cc = sigmoidf_(iv) * fast_tanh(gv);
        const float hh = sigmoidf_(ov) * fast_tanh(cc);
        query[(size_t)row * 256 + col]       = hh;   // h half
        query[(size_t)row * 256 + 128 + col] = cc;   // c half
    }
}

// ---------------------------------------------------------------------------
// Kernel B: s2 = query(8192x256) @ w2(256x128) + b2.  One wave per 16x16 tile.
// ---------------------------------------------------------------------------
__global__ __launch_bounds__(128) void attn_gemm_kernel(
    const float* __restrict__ query, const float* __restrict__ W2,
    const float* __restrict__ b2, float* __restrict__ s2)
{
    const int wave = (blockIdx.x << 2) + (threadIdx.x >> 5);
    const int lane = threadIdx.x & 31;
    const int lm   = lane & 15;
    const int kh   = (lane >> 4) << 1;

    const int ct = wave & 7;               // 8 tiles cover 128 output cols
    const int m0 = (wave >> 3) << 4;
    const int c0 = ct << 4;

    v8f acc = {};
    const float* arow = query + (size_t)(m0 + lm) * 256;
    for (int kk = 0; kk < 256; kk += 4) {
        const int k0 = kk + kh;
        v2f a; a.x = arow[k0]; a.y = arow[k0 + 1];
        v2f b;
        b.x = W2[(size_t)k0 * 128 + c0 + lm];
        b.y = W2[(size_t)(k0 + 1) * 128 + c0 + lm];
        acc = __builtin_amdgcn_wmma_f32_16x16x4_f32(false, a, false, b, (short)0, acc, false, false);
    }

    const int rbase = m0 + ((lane >> 4) << 3);
    const int col   = c0 + lm;
    const float bb  = b2[col];
#pragma unroll
    for (int r = 0; r < 8; ++r) {
        s2[(size_t)(rbase + r) * 128 + col] = acc[r] + bb;
    }
}

// ---------------------------------------------------------------------------
// Kernel C: fused score/einsum/softmax/output. One 128-thread block per
// (b,t) row; thread n computes e[n] = sum_k v[k]*tanh(x*w1k[k]+w1b[k]+s2[k]),
// then a 4-wave softmax over n, then out = x * alpha.  Never materializes
// the (B,T,N,T) score tensor (saves ~1 GB of HBM traffic).  Inner loop is
// fma + v_tanh_f32 (TRANS) + fma per k -> TRANS/VALU co-execution.
// ---------------------------------------------------------------------------
__global__ __launch_bounds__(128) void attn_softmax_kernel(
    const float* __restrict__ x,   const float* __restrict__ s2,
    const float* __restrict__ w1k, const float* __restrict__ w1b,
    const float* __restrict__ v,   const float* __restrict__ vb,
    float* __restrict__ out)
{
    __shared__ float s_s2[128];
    __shared__ float s_w1k[128];
    __shared__ float s_v[128];
    __shared__ float s_red[4];

    const int row = blockIdx.x;
    const int tid = threadIdx.x;

    s_s2[tid]  = s2[(size_t)row * 128 + tid] + w1b[tid];  // fold w1_bias once
    s_w1k[tid] = w1k[tid];
    s_v[tid]   = v[tid];
    __syncthreads();

    const float xv = x[(size_t)row * 128 + tid];
    float acc = 0.0f;
#pragma unroll 8
    for (int k = 0; k < 128; ++k) {
        acc = __builtin_fmaf(s_v[k],
                             fast_tanh(__builtin_fmaf(xv, s_w1k[k], s_s2[k])),
                             acc);
    }
    acc += vb[0];   // constant shift (cancels in softmax, kept for fidelity)

    const int lane = tid & 31;
    const int wid  = tid >> 5;

    // max over 128 lanes: intra-wave shfl (wave32) + LDS across 4 waves
    float m = acc;
#pragma unroll
    for (int off = 16; off >= 1; off >>= 1)
        m = fmaxf(m, __shfl_xor(m, off, 32));
    if (lane == 0) s_red[wid] = m;
    __syncthreads();
    m = fmaxf(fmaxf(s_red[0], s_red[1]), fmaxf(s_red[2], s_red[3]));
    __syncthreads();

    const float ex = __expf(acc - m);
    float s = ex;
#pragma unroll
    for (int off = 16; off >= 1; off >>= 1)
        s += __shfl_xor(s, off, 32);
    if (lane == 0) s_red[wid] = s;
    __syncthreads();
    s = s_red[0] + s_red[1] + s_red[2] + s_red[3];

    out[(size_t)row * 128 + tid] = xv * ex / s;
}

// ---------------------------------------------------------------------------
extern "C" void kernel_launch(void* const* d_in, const int* in_sizes, int n_in,
                              void* d_out, int out_size, void* d_ws, size_t ws_size,
                              hipStream_t stream) {
    const float* x           = (const float*)d_in[0];
    // d_in[1] hidden_state, d_in[2] cell_state: unused (reference ignores them)
    const float* lstm_kernel = (const float*)d_in[3];   // (128, 512)
    const float* lstm_bias   = (const float*)d_in[4];   // (512)
    const float* w1_kernel   = (const float*)d_in[5];   // (1, 128)
    const float* w1_bias     = (const float*)d_in[6];   // (128)
    const float* w2_kernel   = (const float*)d_in[7];   // (256, 128)
    const float* w2_bias     = (const float*)d_in[8];   // (128)
    const float* v_kernel    = (const float*)d_in[9];   // (128, 1)
    const float* v_bias      = (const float*)d_in[10];  // (1)
    float* out = (float*)d_out;

    float* query = (float*)d_ws;             // 8192*256 f32 = 8 MB
    float* s2    = query + (size_t)BT * 256; // 8192*128 f32 = 4 MB

    // 4096 tiles, 4 waves/block -> 1024 blocks
    lstm_gemm_kernel<<<1024, 128, 0, stream>>>(x, lstm_kernel, lstm_bias, query);
    attn_gemm_kernel<<<1024, 128, 0, stream>>>(query, w2_kernel, w2_bias, s2);
    attn_softmax_kernel<<<BT, 128, 0, stream>>>(x, s2, w1_kernel, w1_bias,
                                                v_kernel, v_bias, out);
}